// EntropicOT_56573309223308
// MI455X (gfx1250) — compile-verified
//
#include <hip/hip_runtime.h>
#include <math.h>

// ---------------- problem constants ----------------
constexpr int   kN       = 8192;
constexpr int   kD       = 64;
constexpr float kEpsInv  = 10.0f;     // 1/EPSILON
constexpr int   kIters   = 100;
constexpr int   kRSplit  = 64;        // row-splits for the column-LSE pass

typedef __attribute__((ext_vector_type(2))) float v2f;
typedef __attribute__((ext_vector_type(4))) float v4f;
typedef __attribute__((ext_vector_type(8))) float v8f;

// ---------------- online-LSE helpers ----------------
__device__ __forceinline__ void lse_push(float& m, float& s, float x) {
  if (x > m) { s = fmaf(s, __expf(m - x), 1.0f); m = x; }
  else       { s += __expf(x - m); }
}
__device__ __forceinline__ void lse_merge(float& m, float& s, float om, float os) {
  float nm = fmaxf(m, om);
  s = s * __expf(m - nm) + os * __expf(om - nm);
  m = nm;
}

// ---------------- row squared norms ----------------
__global__ void sqnorm_kernel(const float* __restrict__ src,
                              const float* __restrict__ tgt,
                              float* __restrict__ x2,
                              float* __restrict__ y2) {
  int t = blockIdx.x * blockDim.x + threadIdx.x;
  const float* base = (t < kN) ? (src + (size_t)t * kD)
                               : (tgt + (size_t)(t - kN) * kD);
  float acc = 0.0f;
#pragma unroll
  for (int k = 0; k < kD; k += 4) {
    v4f q = *(const v4f*)(base + k);
    acc = fmaf(q.x, q.x, acc);
    acc = fmaf(q.y, q.y, acc);
    acc = fmaf(q.z, q.z, acc);
    acc = fmaf(q.w, q.w, acc);
  }
  if (t < kN) x2[t] = acc; else y2[t - kN] = acc;
}

// ---------------- cost matrix via exact-fp32 WMMA ----------------
// One wave per 16x16 tile of C. K=64 -> 16x V_WMMA_F32_16X16X4_F32.
// A lane layout (16x4 f32): lane L holds A[M=L&15][K=2*(L>>4)+{0,1}]
// B lane layout (4x16 f32): lane L holds B[K=2*(L>>4)+{0,1}][N=L&15] = target[N][K]
// C/D layout: VGPR r, lane L -> (M = r + 8*(L>>4), N = L&15)
__global__ void cost_wmma_kernel(const float* __restrict__ src,
                                 const float* __restrict__ tgt,
                                 const float* __restrict__ x2,
                                 const float* __restrict__ y2,
                                 float* __restrict__ Cout) {
  int wave = (blockIdx.x * blockDim.x + threadIdx.x) >> 5;
  int lane = threadIdx.x & 31;
  int ti   = wave >> 9;            // tile row   (kN/16 = 512 tiles)
  int tj   = wave & 511;           // tile col
  int m    = lane & 15;
  int hi   = lane >> 4;

  const float* arow = src + (size_t)(ti * 16 + m) * kD + 2 * hi;
  const float* brow = tgt + (size_t)(tj * 16 + m) * kD + 2 * hi;

  v8f acc = {};
#pragma unroll
  for (int k0 = 0; k0 < kD; k0 += 4) {
    v2f a = *(const v2f*)(arow + k0);   // 8B-aligned (k0 and 2*hi even)
    v2f b = *(const v2f*)(brow + k0);
    acc = __builtin_amdgcn_wmma_f32_16x16x4_f32(
        /*neg_a=*/false, a, /*neg_b=*/false, b,
        /*c_mod=*/(short)0, acc, /*reuse_a=*/false, /*reuse_b=*/false);
  }

  int   jg = tj * 16 + m;
  float yj = y2[jg];
#pragma unroll
  for (int r = 0; r < 8; ++r) {
    int ig = ti * 16 + r + 8 * hi;
    // regular (RT) store: C is re-streamed 200x -> keep it L2-resident
    Cout[(size_t)ig * kN + jg] = x2[ig] + yj - 2.0f * acc[r];
  }
}

// ---------------- init potentials ----------------
__global__ void init_uv_kernel(float* __restrict__ u, float* __restrict__ v) {
  int t = blockIdx.x * blockDim.x + threadIdx.x;
  u[t] = 0.0f;
  v[t] = 0.0f;
}

// ---------------- column-LSE pass (reduce over rows i) ----------------
// grid(kN/1024, kRSplit); each thread owns 4 consecutive columns (b128 loads),
// 4 independent online-LSE chains for VALU ILP. One v_exp_f32 per element.
__global__ void col_lse_partial_kernel(const float* __restrict__ Cmat,
                                       const float* __restrict__ v,
                                       float* __restrict__ pm,
                                       float* __restrict__ ps) {
  int j4 = (blockIdx.x * blockDim.x + threadIdx.x) * 4;   // first of 4 columns
  int rb = blockIdx.y * (kN / kRSplit);
  int re = rb + (kN / kRSplit);
  float m0 = -3.0e38f, m1 = -3.0e38f, m2 = -3.0e38f, m3 = -3.0e38f;
  float s0 = 0.0f, s1 = 0.0f, s2 = 0.0f, s3 = 0.0f;
  for (int i = rb; i < re; ++i) {
    const float* rowp = Cmat + (size_t)i * kN + j4;
    if ((threadIdx.x & 7) == 0)                          // one prefetch per 128B line
      __builtin_prefetch(rowp + 4 * (size_t)kN, 0, 1);   // 4 rows ahead
    v4f c  = *(const v4f*)rowp;
    float vi = v[i];                                     // uniform -> scalar load
    lse_push(m0, s0, fmaf(-kEpsInv, c.x, vi));
    lse_push(m1, s1, fmaf(-kEpsInv, c.y, vi));
    lse_push(m2, s2, fmaf(-kEpsInv, c.z, vi));
    lse_push(m3, s3, fmaf(-kEpsInv, c.w, vi));
  }
  size_t o = (size_t)blockIdx.y * kN + j4;
  v4f om = {m0, m1, m2, m3};
  v4f os = {s0, s1, s2, s3};
  *(v4f*)(pm + o) = om;
  *(v4f*)(ps + o) = os;
}

// combine kRSplit partials -> u[j] = exp(log_a - LSE)
__global__ void col_combine_kernel(const float* __restrict__ pm,
                                   const float* __restrict__ ps,
                                   float* __restrict__ u) {
  int j = blockIdx.x * blockDim.x + threadIdx.x;
  float mx = -3.0e38f, sm = 0.0f;
#pragma unroll
  for (int r = 0; r < kRSplit; ++r)
    lse_merge(mx, sm, pm[(size_t)r * kN + j], ps[(size_t)r * kN + j]);
  const float log_a = __logf(1.0f / kN + 1e-8f);
  u[j] = __expf(log_a - (mx + __logf(sm)));
}

// ---------------- row-LSE pass (reduce over cols j) ----------------
// One wave32 per row; each lane owns 4 consecutive columns per step (b128
// loads, 512B per wave-step), 4 independent LSE chains, then butterfly merge.
__global__ void row_lse_kernel(const float* __restrict__ Cmat,
                               const float* __restrict__ u,
                               float* __restrict__ v) {
  int gw   = (blockIdx.x * blockDim.x + threadIdx.x) >> 5;
  int lane = threadIdx.x & 31;
  const float* row = Cmat + (size_t)gw * kN;
  float m0 = -3.0e38f, m1 = -3.0e38f, m2 = -3.0e38f, m3 = -3.0e38f;
  float s0 = 0.0f, s1 = 0.0f, s2 = 0.0f, s3 = 0.0f;
  for (int j = lane * 4; j < kN; j += 128) {
    if (lane == 0) __builtin_prefetch(row + j + 2048, 0, 1);  // 8KB ahead
    v4f c  = *(const v4f*)(row + j);
    v4f uu = *(const v4f*)(u + j);
    lse_push(m0, s0, fmaf(-kEpsInv, c.x, uu.x));
    lse_push(m1, s1, fmaf(-kEpsInv, c.y, uu.y));
    lse_push(m2, s2, fmaf(-kEpsInv, c.z, uu.z));
    lse_push(m3, s3, fmaf(-kEpsInv, c.w, uu.w));
  }
  lse_merge(m0, s0, m1, s1);
  lse_merge(m2, s2, m3, s3);
  lse_merge(m0, s0, m2, s2);
#pragma unroll
  for (int off = 16; off; off >>= 1) {
    float om = __shfl_xor(m0, off, 32);
    float os = __shfl_xor(s0, off, 32);
    lse_merge(m0, s0, om, os);
  }
  if (lane == 0) {
    const float log_b = __logf(1.0f / kN + 1e-8f);
    v[gw] = __expf(log_b - (m0 + __logf(s0)));
  }
}

// ---------------- final transport plan ----------------
// P[i,j] = exp(-C/eps + v[i] + u[j]); float4 per thread; NT stores so the
// 256MB write-once plan does not evict the L2-resident C matrix.
__global__ void plan_kernel(const float* __restrict__ Cmat,
                            const float* __restrict__ u,
                            const float* __restrict__ v,
                            float* __restrict__ P) {
  size_t t  = (size_t)blockIdx.x * blockDim.x + threadIdx.x;  // float4 index
  int    j4 = (int)(t % (kN / 4));
  int    i  = (int)(t / (kN / 4));
  v4f c  = *(const v4f*)(Cmat + (size_t)i * kN + j4 * 4);
  v4f uu = *(const v4f*)(u + j4 * 4);
  float vi = v[i];
  v4f p;
  p.x = __expf(fmaf(-kEpsInv, c.x, vi + uu.x));
  p.y = __expf(fmaf(-kEpsInv, c.y, vi + uu.y));
  p.z = __expf(fmaf(-kEpsInv, c.z, vi + uu.z));
  p.w = __expf(fmaf(-kEpsInv, c.w, vi + uu.w));
  __builtin_nontemporal_store(p, (v4f*)(P + (size_t)i * kN + j4 * 4));
}

// ---------------- host-side launch ----------------
extern "C" void kernel_launch(void* const* d_in, const int* in_sizes, int n_in,
                              void* d_out, int out_size, void* d_ws, size_t ws_size,
                              hipStream_t stream) {
  (void)in_sizes; (void)n_in; (void)out_size; (void)ws_size;
  const float* src = (const float*)d_in[0];
  const float* tgt = (const float*)d_in[1];

  // d_out = [transport_plan (N*N) | C (N*N)]  (tuple return order)
  float* plan = (float*)d_out;
  float* Cmat = plan + (size_t)kN * kN;

  // workspace layout (floats): x2 | y2 | u | v | pm | ps   (~4.2 MB)
  float* ws = (float*)d_ws;
  float* x2 = ws;
  float* y2 = ws + kN;
  float* u  = ws + 2 * kN;
  float* v  = ws + 3 * kN;
  float* pm = ws + 4 * kN;
  float* ps = pm + (size_t)kRSplit * kN;

  sqnorm_kernel<<<2 * kN / 256, 256, 0, stream>>>(src, tgt, x2, y2);

  // (kN/16)^2 = 262144 tiles, 8 waves/block -> 32768 blocks
  cost_wmma_kernel<<<(kN / 16) * (kN / 16) / 8, 256, 0, stream>>>(src, tgt, x2, y2, Cmat);

  init_uv_kernel<<<kN / 256, 256, 0, stream>>>(u, v);

  for (int it = 0; it < kIters; ++it) {
    col_lse_partial_kernel<<<dim3(kN / 1024, kRSplit), 256, 0, stream>>>(Cmat, v, pm, ps);
    col_combine_kernel<<<kN / 256, 256, 0, stream>>>(pm, ps, u);
    row_lse_kernel<<<kN * 32 / 256, 256, 0, stream>>>(Cmat, u, v);
  }

  plan_kernel<<<(int)((size_t)kN * kN / 4 / 256), 256, 0, stream>>>(Cmat, u, v, plan);
}